// STIMGAT_37735582663325
// MI455X (gfx1250) — compile-verified
//
#include <hip/hip_runtime.h>
#include <math.h>

// ---------------------------------------------------------------------------
// STAGATE GAT forward for MI455X (gfx1250, wave32).
//  * GEMMs: native fp32 WMMA (v_wmma_f32_16x16x4_f32). Each block stages one
//    K x 64 weight panel into LDS via the Tensor Data Mover (tensor_load_to_lds
//    + s_wait_tensorcnt), with TDM row padding (+4 dwords) for bank-conflict-
//    free ds_load_b64 fragment reads. Each wave computes a 16x64 output strip
//    (4 accumulators), amortizing the A fragment 4x.
//  * Edge softmax: L2-resident atomics (hp tables are 51.2MB < 192MB L2).
//    Attention weights are tied between conv1 and conv3 -> computed once.
// ---------------------------------------------------------------------------

typedef float v2f __attribute__((ext_vector_type(2)));
typedef float v8f __attribute__((ext_vector_type(8)));
typedef unsigned int v4u __attribute__((ext_vector_type(4)));
typedef int v4i __attribute__((ext_vector_type(4)));
typedef int v8i __attribute__((ext_vector_type(8)));

#define IN_DIM 512
#define HID    256
#define OUTD   64
#define NEG_SLOPE 0.2f
#define LDS_PAD 4   // dwords of padding per LDS panel row (TDM pad_amount code 3)

#if defined(__HIP_DEVICE_COMPILE__) && __has_builtin(__builtin_amdgcn_tensor_load_to_lds)
#define HAVE_TDM 1
#else
#define HAVE_TDM 0
#endif

__device__ __forceinline__ int orderedFromFloat(float f) {
  int i = __float_as_int(f);
  return i >= 0 ? i : (i ^ 0x7FFFFFFF);
}
__device__ __forceinline__ float floatFromOrdered(int i) {
  return __int_as_float(i >= 0 ? i : (i ^ 0x7FFFFFFF));
}

// ---------------- fp32 WMMA GEMM: C[M x Nc] = A[M x K] @ B ------------------
// TRANSB=0: B is [K x Nc] row-major.  TRANSB=1: B is [Nc x K] row-major (B^T).
// Grid: x = strips of 8 M-tiles (8 waves/block, 16 rows each), y = Nc/64 panel.
// LDS panel: TRANSB -> [64][K+PAD] (b-frag = one ds_load_b64),
//            else   -> [K][64+PAD] (two ds_load_b32, consecutive banks).
template <bool TRANSB>
__global__ void k_gemm_wmma(const float* __restrict__ A,
                            const float* __restrict__ B,
                            float* __restrict__ C,
                            int M, int Nc, int K) {
  extern __shared__ float Bs[];
  const int colBase = blockIdx.y * 64;
  const int rowLen = TRANSB ? K : 64;   // contiguous dwords per panel row
  const int rows   = TRANSB ? 64 : K;
  const int ldb    = rowLen + LDS_PAD;

#if HAVE_TDM
  if ((threadIdx.x >> 5) == 0) {   // one wave issues the TDM DMA
    const float* tile = TRANSB ? (B + (size_t)colBase * K) : (B + colBase);
    const unsigned long long strideEl =
        TRANSB ? (unsigned long long)K : (unsigned long long)Nc;
    const unsigned long long ga = (unsigned long long)(size_t)tile;
    const unsigned d0 = (unsigned)rowLen;          // tile/tensor dim0
    const unsigned d1 = (unsigned)rows;            // tile/tensor dim1
    // pad_interval code c: pads after 2^(c+1) dwords == rowLen (64->5, 256->7)
    const unsigned ivl = (unsigned)(__builtin_ctz((unsigned)rowLen) - 1);

    v4u g0;
    g0[0] = 1u;                                    // count=1 (valid descriptor)
    g0[1] = (unsigned)(size_t)Bs;                  // lds_addr (low 32 of flat)
    g0[2] = (unsigned)(ga & 0xFFFFFFFFull);        // global_addr[31:0]
    g0[3] = (unsigned)((ga >> 32) & 0x1FFFFFFull)  // global_addr[56:32]
            | (2u << 30);                          // type = 2 ("image")

    v8i g1 = {};
    g1[0] = (int)((2u << 16)                        // data_size = 4B
                  | (1u << 20)                      // pad_enable
                  | (ivl << 22)                     // pad_interval
                  | (3u << 25));                    // pad_amount = 4 dwords
    g1[1] = (int)((d0 & 0xFFFFu) << 16);            // tensor_dim0[15:0]
    g1[2] = (int)(((d0 >> 16) & 0xFFFFu) | ((d1 & 0xFFFFu) << 16)); // td0 hi | td1 lo
    g1[3] = (int)(((d1 >> 16) & 0xFFFFu) | ((d0 & 0xFFFFu) << 16)); // td1 hi | tile_dim0
    g1[4] = (int)(d1 & 0xFFFFu);                    // tile_dim1 (tile_dim2 = 0)
    g1[5] = (int)(unsigned)(strideEl & 0xFFFFFFFFull);        // dim0_stride lo
    g1[6] = (int)(unsigned)((strideEl >> 32) & 0xFFFFull);    // dim0_stride hi
    g1[7] = 0;

    const v4i gz = {0, 0, 0, 0};
#if defined(__clang_major__) && (__clang_major__ >= 23)
    const v8i gz8 = {};
    __builtin_amdgcn_tensor_load_to_lds(g0, g1, gz, gz, gz8, 0);
#else
    __builtin_amdgcn_tensor_load_to_lds(g0, g1, gz, gz, 0);
#endif
    __builtin_amdgcn_s_wait_tensorcnt(0);
  }
  __syncthreads();
#else
  for (int idx = threadIdx.x; idx < rows * rowLen; idx += blockDim.x) {
    const int r = idx / rowLen, c = idx - r * rowLen;
    Bs[r * ldb + c] = TRANSB ? B[(size_t)(colBase + r) * K + c]
                             : B[(size_t)r * Nc + colBase + c];
  }
  __syncthreads();
#endif

  const int tileM = blockIdx.x * (blockDim.x >> 5) + (threadIdx.x >> 5);
  if (tileM * 16 >= M) return;

  const int lane = threadIdx.x & 31;
  const int half = lane >> 4;   // 0: k+0/k+1, 1: k+2/k+3 (A & B symmetric)
  const int l16  = lane & 15;
  const float* Arow = A + (size_t)(tileM * 16 + l16) * K;

  v8f acc[4] = {{}, {}, {}, {}};
#pragma unroll 2
  for (int k = 0; k < K; k += 4) {
    const int ka = k + 2 * half;
    const v2f a = *(const v2f*)(Arow + ka);
#pragma unroll
    for (int j = 0; j < 4; ++j) {
      v2f b;
      if (TRANSB) {
        b = *(const v2f*)&Bs[(j * 16 + l16) * ldb + ka];
      } else {
        b.x = Bs[ka * ldb + j * 16 + l16];
        b.y = Bs[(ka + 1) * ldb + j * 16 + l16];
      }
      acc[j] = __builtin_amdgcn_wmma_f32_16x16x4_f32(
          false, a, false, b, (short)0, acc[j], false, false);
    }
  }

  // C/D layout: VGPR r -> row (r + 8*half), col = l16.
  float* Cp = C + (size_t)(tileM * 16 + 8 * half) * Nc + colBase + l16;
#pragma unroll
  for (int j = 0; j < 4; ++j)
#pragma unroll
    for (int r = 0; r < 8; ++r) Cp[(size_t)r * Nc + j * 16] = acc[j][r];
}

// --------------- per-node attention scores: a_s, a_d = hp @ att -------------
__global__ void k_node_dots(const float* __restrict__ hp,
                            const float* __restrict__ att_s,
                            const float* __restrict__ att_d,
                            float* __restrict__ a_s, float* __restrict__ a_d,
                            int N) {
  const int wave = (blockIdx.x * blockDim.x + threadIdx.x) >> 5;
  const int lane = threadIdx.x & 31;
  if (wave >= N) return;
  const float* row = hp + (size_t)wave * HID;
  float s = 0.f, d = 0.f;
#pragma unroll
  for (int c = lane; c < HID; c += 32) {
    const float v = row[c];
    s = fmaf(v, att_s[c], s);
    d = fmaf(v, att_d[c], d);
  }
#pragma unroll
  for (int off = 16; off > 0; off >>= 1) {
    s += __shfl_down(s, off, 32);
    d += __shfl_down(d, off, 32);
  }
  if (lane == 0) { a_s[wave] = s; a_d[wave] = d; }
}

// --------------- init: mEnc=-inf(encoded), denom=0, acc=0 -------------------
__global__ void k_init_prop(int* __restrict__ mEnc, float* __restrict__ denom,
                            float* __restrict__ acc, int N, int total) {
  const int t = blockIdx.x * blockDim.x + threadIdx.x;
  if (t < total) acc[t] = 0.f;
  if (t < N) { mEnc[t] = (int)0x807FFFFF; denom[t] = 0.f; }  // enc(-inf)
}

__global__ void k_zero(float* __restrict__ p, int total) {
  const int t = blockIdx.x * blockDim.x + threadIdx.x;
  if (t < total) p[t] = 0.f;
}

// --------------- edge passes: logit+max, exp+sum, normalize -----------------
__global__ void k_edge_logit_max(const int* __restrict__ src,
                                 const int* __restrict__ dst,
                                 const float* __restrict__ a_s,
                                 const float* __restrict__ a_d,
                                 int* __restrict__ mEnc,
                                 float* __restrict__ ew, int E) {
  const int e = blockIdx.x * blockDim.x + threadIdx.x;
  if (e >= E) return;
  float l = a_s[src[e]] + a_d[dst[e]];
  l = l > 0.f ? l : NEG_SLOPE * l;           // leaky_relu
  ew[e] = l;
  atomicMax(&mEnc[dst[e]], orderedFromFloat(l));
}

__global__ void k_edge_exp(const int* __restrict__ dst,
                           const int* __restrict__ mEnc,
                           float* __restrict__ denom,
                           float* __restrict__ ew, int E) {
  const int e = blockIdx.x * blockDim.x + threadIdx.x;
  if (e >= E) return;
  const int d = dst[e];
  const float ex = __expf(ew[e] - floatFromOrdered(mEnc[d]));
  ew[e] = ex;
  atomicAdd(&denom[d], ex);
}

__global__ void k_edge_norm(const int* __restrict__ dst,
                            const float* __restrict__ denom,
                            float* __restrict__ ew, int E) {
  const int e = blockIdx.x * blockDim.x + threadIdx.x;
  if (e >= E) return;
  ew[e] = ew[e] / (denom[dst[e]] + 1e-16f);
}

// --------------- weighted scatter: out[dst] += hp[src] * w (one wave/edge) --
__global__ void k_edge_scatter(const int* __restrict__ src,
                               const int* __restrict__ dst,
                               const float* __restrict__ w,
                               const float* __restrict__ hp,
                               float* __restrict__ out, int E) {
  const int e = (blockIdx.x * blockDim.x + threadIdx.x) >> 5;
  const int lane = threadIdx.x & 31;
  if (e >= E) return;
  const int s = src[e], d = dst[e];
  const float wt = w[e];
  const float* hrow = hp + (size_t)s * HID;
  float* orow = out + (size_t)d * HID;
#pragma unroll
  for (int c = lane; c < HID; c += 32)
    atomicAdd(&orow[c], hrow[c] * wt);   // L2-resident float atomics
}

// --------------- ELU in place ----------------------------------------------
__global__ void k_elu(float* __restrict__ x, int total) {
  const int t = blockIdx.x * blockDim.x + threadIdx.x;
  if (t >= total) return;
  const float v = x[t];
  x[t] = v > 0.f ? v : expm1f(v);
}

// ---------------------------------------------------------------------------
static inline size_t panel_bytes(bool transB, int K) {
  return (size_t)(transB ? 64 * (K + LDS_PAD) : K * (64 + LDS_PAD)) * sizeof(float);
}

extern "C" void kernel_launch(void* const* d_in, const int* in_sizes, int n_in,
                              void* d_out, int out_size, void* d_ws, size_t ws_size,
                              hipStream_t stream) {
  const float* feat  = (const float*)d_in[0];   // [N, 512]
  const float* W1    = (const float*)d_in[1];   // [512, 256]
  const float* W2    = (const float*)d_in[2];   // [256, 64]
  const float* att_s = (const float*)d_in[3];   // [256]
  const float* att_d = (const float*)d_in[4];   // [256]
  const int*   eidx  = (const int*)d_in[5];     // [2, E]

  const int N = in_sizes[0] / IN_DIM;
  const int E = in_sizes[5] / 2;
  const int* src = eidx;
  const int* dst = eidx + E;

  // workspace layout (floats): P[N*HID] | Q[N*HID] | a_s[N] | a_d[N] |
  //                            mEnc[N](int) | denom[N] | ew[E]
  float* P     = (float*)d_ws;                  // hp1, later hp3
  float* Q     = P + (size_t)N * HID;           // h1 acc, later h3 acc
  float* a_s   = Q + (size_t)N * HID;
  float* a_d   = a_s + N;
  int*   mEnc  = (int*)(a_d + N);
  float* denom = (float*)(mEnc + N);
  float* ew    = denom + N;

  float* h2 = (float*)d_out;                    // [N, 64]
  float* h4 = h2 + (size_t)N * OUTD;            // [N, 512]

  const dim3 blk(256);
  const int NH = N * HID;
  const int mStrips = ((N / 16) + 7) / 8;       // 8 waves (M-tiles) per block

  // 1) hp1 = feat @ W1                                (WMMA f32)
  k_gemm_wmma<false><<<dim3(mStrips, HID / 64), blk, panel_bytes(false, IN_DIM),
                       stream>>>(feat, W1, P, N, HID, IN_DIM);

  // 2) a_s = hp1 @ att_src, a_d = hp1 @ att_dst
  k_node_dots<<<dim3((N + 7) / 8), blk, 0, stream>>>(P, att_s, att_d, a_s, a_d, N);

  // 3) edge softmax weights (shared by conv1 and conv3 — tied attention)
  k_init_prop<<<dim3((NH + 255) / 256), blk, 0, stream>>>(mEnc, denom, Q, N, NH);
  k_edge_logit_max<<<dim3((E + 255) / 256), blk, 0, stream>>>(src, dst, a_s, a_d, mEnc, ew, E);
  k_edge_exp<<<dim3((E + 255) / 256), blk, 0, stream>>>(dst, mEnc, denom, ew, E);
  k_edge_norm<<<dim3((E + 255) / 256), blk, 0, stream>>>(dst, denom, ew, E);

  // 4) conv1 propagate: Q = scatter(hp1 * w), then ELU -> h1
  k_edge_scatter<<<dim3((E + 7) / 8), blk, 0, stream>>>(src, dst, ew, P, Q, E);
  k_elu<<<dim3((NH + 255) / 256), blk, 0, stream>>>(Q, NH);

  // 5) h2 = h1 @ W2                                   (WMMA f32) -> d_out
  k_gemm_wmma<false><<<dim3(mStrips, OUTD / 64), blk, panel_bytes(false, HID),
                       stream>>>(Q, W2, h2, N, OUTD, HID);

  // 6) hp3 = h2 @ W2^T                                (WMMA f32, transB)
  k_gemm_wmma<true><<<dim3(mStrips, HID / 64), blk, panel_bytes(true, OUTD),
                      stream>>>(h2, W2, P, N, HID, OUTD);

  // 7) conv3 propagate with the same weights w, then ELU -> h3
  k_zero<<<dim3((NH + 255) / 256), blk, 0, stream>>>(Q, NH);
  k_edge_scatter<<<dim3((E + 7) / 8), blk, 0, stream>>>(src, dst, ew, P, Q, E);
  k_elu<<<dim3((NH + 255) / 256), blk, 0, stream>>>(Q, NH);

  // 8) h4 = h3 @ W1^T                                 (WMMA f32, transB)
  k_gemm_wmma<true><<<dim3(mStrips, IN_DIM / 64), blk, panel_bytes(true, HID),
                      stream>>>(Q, W1, h4, N, IN_DIM, HID);
}